// DownSampler_16664473108712
// MI455X (gfx1250) — compile-verified
//
#include <hip/hip_runtime.h>
#include <cstdint>

// Problem constants (fixed by the reference harness)
constexpr int B_   = 8;
constexpr int C_   = 3;
constexpr int H_   = 1024;
constexpr int W_   = 1024;
constexpr int HO_  = 512;
constexpr int WO_  = 512;
constexpr int K2_  = 9;
constexpr int TILE_W = 256;
constexpr int PLANES = 3 * K2_;               // kernels(9) + offsets_h(9) + offsets_v(9)
constexpr int CHUNKS = PLANES * (TILE_W / 4); // 16-byte chunks to stage = 1728

// 2-float vector loadable at 4-byte alignment (hardware DWORD-alignment mode
// allows dword-aligned b64 loads; lets the compiler fuse the x-neighbor pair
// into a single global_load_b64).
typedef float f2 __attribute__((ext_vector_type(2)));
struct __attribute__((packed, aligned(4))) f2u { f2 v; };

// Map a clipped *padded* index (pad=1, range [0, H+1]) to the original image
// index under 'reflect' padding: padded[0]=img[1], padded[H+1]=img[H-2].
__device__ __forceinline__ int reflect1(int ip) {
  int io = ip - 1;                                // io in [-1, H]
  io = io < 0 ? -io : io;                         // -1 -> 1
  io = io > (H_ - 1) ? (2 * (H_ - 1) - io) : io;  // H -> H-2
  return io;
}

__global__ __launch_bounds__(256)
void adaptive_downsample_kernel(const float* __restrict__ img,
                                const float* __restrict__ kern,
                                const float* __restrict__ offh,
                                const float* __restrict__ offv,
                                const float* __restrict__ offset_unit,
                                float* __restrict__ out) {
  __shared__ __align__(16) float smem[PLANES * TILE_W];   // 27 KB

  const int tid = threadIdx.x;
  const int w0  = blockIdx.x * TILE_W;
  const int h   = blockIdx.y;
  const int b   = blockIdx.z;

  // --- Stage 27 per-tap planes (kernels, offsets_h, offsets_v for this row
  // chunk) into LDS via async DMA: 7 x b128 per thread, tracked by ASYNCcnt,
  // overlapped with the address math / prefetch below.
  const unsigned lds_base = (unsigned)(uintptr_t)(void*)smem;
  const long long hw = (long long)HO_ * WO_;
  const long long rowbase = (long long)b * K2_ * hw + (long long)h * WO_ + w0;

#pragma unroll
  for (int it = 0; it < 7; ++it) {
    int f = it * 256 + tid;
    if (f < CHUNKS) {
      int plane = f >> 6;          // 0..26
      int chunk = f & 63;          // 16B chunk within plane (64 per plane)
      int arr   = plane / 9;       // 0=kernels, 1=offsets_h, 2=offsets_v
      int k     = plane - arr * 9;
      const float* basep = (arr == 0) ? kern : (arr == 1 ? offh : offv);
      const float* g = basep + (rowbase + (long long)k * hw + chunk * 4);
      unsigned lds_addr = lds_base + (unsigned)(f * 16);
      asm volatile("global_load_async_to_lds_b128 %0, %1, off"
                   :: "v"(lds_addr),
                      "v"((unsigned long long)(uintptr_t)g)
                   : "memory");
    }
  }

  const int w = w0 + tid;
  const float* imgb = img + (size_t)b * C_ * (size_t)(H_ * W_);

  // Warm the gather window (whole 100MB image fits in 192MB L2).
#pragma unroll
  for (int c = 0; c < C_; ++c)
    __builtin_prefetch(imgb + (size_t)c * (H_ * W_) + (size_t)(2 * h) * W_ + 2 * w, 0, 3);

  const float ou = *offset_unit;   // uniform scalar load

  asm volatile("s_wait_asynccnt 0" ::: "memory");
  __syncthreads();

  float acc0 = 0.f, acc1 = 0.f, acc2 = 0.f;
  const float* p0 = imgb;
  const float* p1 = imgb + (H_ * W_);
  const float* p2 = imgb + 2 * (H_ * W_);

#pragma unroll
  for (int k = 0; k < K2_; ++k) {
    const int ky = k / 3, kx = k % 3;
    float wk = smem[k * TILE_W + tid];
    float oh = smem[(9 + k) * TILE_W + tid];
    float ov = smem[(18 + k) * TILE_W + tid];

    // p = (o + 0.5)/out * in + k_off + offset*unit - 0.5  (padded coords)
    float px = (float)(2 * w + 1 + kx) + oh * ou - 0.5f;
    float py = (float)(2 * h + 1 + ky) + ov * ou - 0.5f;
    float fx = floorf(px), fy = floorf(py);
    float a  = px - fx;
    float bt = py - fy;

    int xL = (int)fx; xL = xL < 0 ? 0 : (xL > W_ + 1 ? W_ + 1 : xL);
    int yT = (int)fy; yT = yT < 0 ? 0 : (yT > H_ + 1 ? H_ + 1 : yT);
    int xR = (xL + 1 > W_ + 1) ? W_ + 1 : xL + 1;
    int yB = (yT + 1 > H_ + 1) ? H_ + 1 : yT + 1;

    int xl = reflect1(xL), xr = reflect1(xR);
    int yt = reflect1(yT), yb = reflect1(yB);

    // After clip+reflect the x-neighbors are always the pair {m, m+1},
    // m = min(xl, xr) in [0, 1022]  (interior: (m, m+1); left edge: (1,0);
    // right edge: (1023,1022) or (1022,1022)).  One b64 load covers both.
    int m = xl < xr ? xl : xr;
    bool xl_lo = (xl == m);
    bool xr_lo = (xr == m);

    float wtl = (1.f - a) * (1.f - bt);
    float wtr = a * (1.f - bt);
    float wbl = (1.f - a) * bt;
    float wbr = a * bt;

    int rT = yt * W_ + m;
    int rB = yb * W_ + m;

    f2 t0 = ((const f2u*)(p0 + rT))->v;
    f2 b0 = ((const f2u*)(p0 + rB))->v;
    f2 t1 = ((const f2u*)(p1 + rT))->v;
    f2 b1 = ((const f2u*)(p1 + rB))->v;
    f2 t2 = ((const f2u*)(p2 + rT))->v;
    f2 b2 = ((const f2u*)(p2 + rB))->v;

    float v0 = wtl * (xl_lo ? t0.x : t0.y) + wtr * (xr_lo ? t0.x : t0.y)
             + wbl * (xl_lo ? b0.x : b0.y) + wbr * (xr_lo ? b0.x : b0.y);
    float v1 = wtl * (xl_lo ? t1.x : t1.y) + wtr * (xr_lo ? t1.x : t1.y)
             + wbl * (xl_lo ? b1.x : b1.y) + wbr * (xr_lo ? b1.x : b1.y);
    float v2 = wtl * (xl_lo ? t2.x : t2.y) + wtr * (xr_lo ? t2.x : t2.y)
             + wbl * (xl_lo ? b2.x : b2.y) + wbr * (xr_lo ? b2.x : b2.y);

    acc0 += wk * v0;
    acc1 += wk * v1;
    acc2 += wk * v2;
  }

  size_t ohw = (size_t)HO_ * WO_;
  size_t ob  = (size_t)b * C_ * ohw + (size_t)h * WO_ + w;
  out[ob]           = acc0;
  out[ob + ohw]     = acc1;
  out[ob + 2 * ohw] = acc2;
}

extern "C" void kernel_launch(void* const* d_in, const int* in_sizes, int n_in,
                              void* d_out, int out_size, void* d_ws, size_t ws_size,
                              hipStream_t stream) {
  (void)in_sizes; (void)n_in; (void)out_size; (void)d_ws; (void)ws_size;
  const float* img  = (const float*)d_in[0];
  const float* kern = (const float*)d_in[1];
  const float* offh = (const float*)d_in[2];
  const float* offv = (const float*)d_in[3];
  const float* ou   = (const float*)d_in[4];
  float* out = (float*)d_out;

  dim3 grid(WO_ / TILE_W, HO_, B_);
  adaptive_downsample_kernel<<<grid, TILE_W, 0, stream>>>(img, kern, offh, offv, ou, out);
}